// MMNet_seg_73272142070068
// MI455X (gfx1250) — compile-verified
//
#include <hip/hip_runtime.h>
#include <cstdint>

typedef __attribute__((ext_vector_type(2))) float v2f;
typedef __attribute__((ext_vector_type(8))) float v8f;

#if defined(__has_builtin)
#if __has_builtin(__builtin_amdgcn_global_load_async_to_lds_b32) && \
    __has_builtin(__builtin_amdgcn_s_wait_asynccnt)
#define HAVE_ASYNC_LDS 1
#endif
#endif
#ifndef HAVE_ASYNC_LDS
#define HAVE_ASYNC_LDS 0
#endif

#if HAVE_ASYNC_LDS
typedef __attribute__((address_space(1))) int g1int;   // global (prints as __device__ int*)
typedef __attribute__((address_space(3))) int l3int;   // LDS
#endif

namespace {

constexpr int   BB   = 2;
constexpr int   NP   = 8192;
constexpr int   KNB  = 16;           // neighbors
constexpr int   NSUB = NP / 4;       // 2048
constexpr int   NCLS = 13;
constexpr int   KN   = KNB * NP;     // 131072
constexpr float EPSV = 1e-5f;
constexpr float LEAK = 0.01f;

// ---------------------------------------------------------------- utilities

__global__ void k_build_xyz(const float* __restrict__ x, float* __restrict__ xyz) {
  int i = blockIdx.x * blockDim.x + threadIdx.x;
  if (i >= BB * NP * 3) return;
  int b = i / (3 * NP);
  int f = i % (3 * NP);                       // replicate reshape: slice [3,N] flat
  xyz[i] = x[(size_t)b * 24 * NP + (size_t)(9 + f / NP) * NP + (f % NP)];
}

__global__ void k_norms(const float* __restrict__ xyz, float* __restrict__ qn) {
  int i = blockIdx.x * blockDim.x + threadIdx.x;
  if (i >= BB * NP) return;
  const float* p = xyz + (size_t)i * 3;
  qn[i] = p[0] * p[0] + p[1] * p[1] + p[2] * p[2];
}

// ---------------------------------------------------------------- KNN (WMMA)
// wave handles a 16-query tile; 16x16 distance tile via V_WMMA_F32_16X16X4_F32.
// C layout: lane L<16 holds column L rows 0..7 (vgpr r -> row r),
//           lane L+16 holds column L rows 8..15.
__global__ __launch_bounds__(128) void k_knn(const float* __restrict__ xyz,
                                             const float* __restrict__ qn,
                                             int* __restrict__ neigh) {
  const int lane = threadIdx.x & 31;
  const int wib  = threadIdx.x >> 5;
  const int wid  = blockIdx.x * 4 + wib;
  const int tilesPerB = NP / 16;
  if (wid >= BB * tilesPerB) return;
  const int b   = wid / tilesPerB;
  const int qt  = wid % tilesPerB;
  const int col = lane & 15;
  const bool hi = lane >= 16;
  const int rowoff = hi ? 8 : 0;
  const int q = qt * 16 + col;
  const float* X  = xyz + (size_t)b * NP * 3;
  const float* QN = qn + (size_t)b * NP;

  // B fragment = query coords (4 x 16): unconditional loads + selects
  const float* qp = X + (size_t)q * 3;
  float qx = qp[0], qy = qp[1], qz = qp[2];
  v2f bfrag;
  bfrag.x = hi ? qz : qx;
  bfrag.y = hi ? 0.f : qy;
  const float myqn = QN[q];

  float bd[KNB]; int bi[KNB];
#pragma unroll
  for (int t = 0; t < KNB; t++) { bd[t] = 3.0e38f; bi[t] = 0x7fffffff; }

  for (int s0 = 0; s0 < NP; s0 += 16) {
    if (s0 + 128 < NP) __builtin_prefetch(&X[(s0 + 128) * 3], 0, 0);
    const int am = s0 + col;                   // A = support coords (16 x 4)
    const float* ap = X + (size_t)am * 3;
    float a0 = ap[0], a1 = ap[1], a2 = ap[2];
    v2f afrag;
    afrag.x = hi ? a2 : a0;
    afrag.y = hi ? 0.f : a1;
    v8f acc = {};
    acc = __builtin_amdgcn_wmma_f32_16x16x4_f32(false, afrag, false, bfrag,
                                                (short)0, acc, false, false);
#pragma unroll
    for (int r = 0; r < 8; r++) {
      const int srow = s0 + rowoff + r;
      float nd = myqn + QN[srow] - 2.0f * acc[r];
      int ni = srow;
      // wave-uniform skip; chain below is self-correcting for non-inserting lanes
      if (__builtin_amdgcn_ballot_w32(nd <= bd[KNB - 1]) != 0u) {
#pragma unroll
        for (int t = 0; t < KNB; t++) {
          // bitwise | & : no short-circuit, no exec-mask branches
          bool sw = (bool)((nd < bd[t]) | ((nd == bd[t]) & (ni < bi[t])));
          float od = bd[t]; int oi = bi[t];
          bd[t] = sw ? nd : od;  bi[t] = sw ? ni : oi;
          nd = sw ? od : nd;     ni = sw ? oi : ni;
        }
      }
    }
  }

  __shared__ float sd[4][32][KNB];
  __shared__ int   si[4][32][KNB];
#pragma unroll
  for (int t = 0; t < KNB; t++) { sd[wib][lane][t] = bd[t]; si[wib][lane][t] = bi[t]; }
  __syncthreads();
  if (!hi) {                                   // merge lo/hi sorted lists
    int a = 0, c = 0;
    size_t out = (size_t)b * NP * KNB + (size_t)q * KNB;
#pragma unroll
    for (int t = 0; t < KNB; t++) {
      float da = sd[wib][lane][a], dc = sd[wib][lane + 16][c];
      int   ia = si[wib][lane][a], ic = si[wib][lane + 16][c];
      bool takeA = (bool)((da < dc) | ((da == dc) & (ia <= ic)));
      neigh[out + t] = takeA ? ia : ic;
      a += takeA ? 1 : 0;
      c += takeA ? 0 : 1;
    }
  }
}

// ---------------------------------------------------------------- FPS (serial)
__global__ __launch_bounds__(1024) void k_fps(const float* __restrict__ xyz,
                                              int* __restrict__ subidx) {
  const int b = blockIdx.x;
  const int tid = threadIdx.x;
  const float* X = xyz + (size_t)b * NP * 3;
  __shared__ float rv[1024];
  __shared__ int   ri[1024];
  __shared__ float lastp[3];
  float px[8], py[8], pz[8], dmin[8];
#pragma unroll
  for (int i = 0; i < 8; i++) {
    int p = i * 1024 + tid;
    px[i] = X[p * 3]; py[i] = X[p * 3 + 1]; pz[i] = X[p * 3 + 2];
    dmin[i] = 1e10f;
  }
  int last = 0;
  for (int s = 0; s < NSUB; s++) {
    if (tid == 0) {
      subidx[(size_t)b * NSUB + s] = last;     // emit previous index (scan quirk)
      lastp[0] = X[last * 3]; lastp[1] = X[last * 3 + 1]; lastp[2] = X[last * 3 + 2];
    }
    __syncthreads();
    const float lx = lastp[0], ly = lastp[1], lz = lastp[2];
    float bestd = -1.f; int besti = 0;
#pragma unroll
    for (int i = 0; i < 8; i++) {
      float dx = px[i] - lx, dy = py[i] - ly, dz = pz[i] - lz;
      float d = dx * dx + dy * dy + dz * dz;
      dmin[i] = fminf(dmin[i], d);
      bool better = dmin[i] > bestd;
      besti = better ? (i * 1024 + tid) : besti;
      bestd = better ? dmin[i] : bestd;
    }
    rv[tid] = bestd; ri[tid] = besti;
    __syncthreads();
    for (int off = 512; off > 0; off >>= 1) {
      if (tid < off) {
        float o = rv[tid + off]; int oi = ri[tid + off];
        bool take = (bool)((o > rv[tid]) | ((o == rv[tid]) & (oi < ri[tid])));
        rv[tid] = take ? o : rv[tid];
        ri[tid] = take ? oi : ri[tid];
      }
      __syncthreads();
    }
    last = ri[0];
    __syncthreads();
  }
}

__global__ void k_subxyz(const float* __restrict__ xyz, const int* __restrict__ subi,
                         float* __restrict__ sxyz) {
  int i = blockIdx.x * blockDim.x + threadIdx.x;
  if (i >= BB * NSUB) return;
  int b = i / NSUB;
  int p = subi[i];
  sxyz[(size_t)i * 3 + 0] = xyz[((size_t)b * NP + p) * 3 + 0];
  sxyz[(size_t)i * 3 + 1] = xyz[((size_t)b * NP + p) * 3 + 1];
  sxyz[(size_t)i * 3 + 2] = xyz[((size_t)b * NP + p) * 3 + 2];
}

__global__ __launch_bounds__(256) void k_interp(const float* __restrict__ xyz,
                                                const float* __restrict__ sxyz,
                                                int* __restrict__ itp) {
  __shared__ float sh[256 * 3];
  int gid = blockIdx.x * 256 + threadIdx.x;     // blocks never straddle batches
  int b = gid / NP;
  const float* Q = xyz + (size_t)gid * 3;
  float qx = Q[0], qy = Q[1], qz = Q[2];
  float best = 3.0e38f; int bidx = 0;
  for (int c0 = 0; c0 < NSUB; c0 += 256) {
    __syncthreads();
    int s = c0 + threadIdx.x;
    const float* g = &sxyz[((size_t)b * NSUB + s) * 3];
#if HAVE_ASYNC_LDS
    {
      unsigned lo = (unsigned)(uintptr_t)(&sh[threadIdx.x * 3]);
      __builtin_amdgcn_global_load_async_to_lds_b32(
          (g1int*)(uintptr_t)(g + 0), (l3int*)(uintptr_t)(lo + 0), 0, 0);
      __builtin_amdgcn_global_load_async_to_lds_b32(
          (g1int*)(uintptr_t)(g + 1), (l3int*)(uintptr_t)(lo + 4), 0, 0);
      __builtin_amdgcn_global_load_async_to_lds_b32(
          (g1int*)(uintptr_t)(g + 2), (l3int*)(uintptr_t)(lo + 8), 0, 0);
      __builtin_amdgcn_s_wait_asynccnt(0);
    }
#else
    sh[threadIdx.x * 3 + 0] = g[0];
    sh[threadIdx.x * 3 + 1] = g[1];
    sh[threadIdx.x * 3 + 2] = g[2];
#endif
    __syncthreads();
    for (int j = 0; j < 256; j++) {
      float dx = qx - sh[j * 3], dy = qy - sh[j * 3 + 1], dz = qz - sh[j * 3 + 2];
      float d = dx * dx + dy * dy + dz * dz;
      bool better = d < best;
      bidx = better ? (c0 + j) : bidx;
      best = better ? d : best;
    }
  }
  itp[gid] = bidx;
}

// ------------------------------------------------------- 1x1 conv via WMMA f32
// out[b, co, m] = sum_ci w[co,ci] * in[b, ci, m] + bias[co]
// wave tile: 16 positions (rows) x 16 out-channels (cols), K stepped by 4.
// Addressing unified by strides: in elem (ci,m) at ci*sKin + m*sMin,
// out elem (co,row) at row*sRow + co*sCo.
__global__ __launch_bounds__(256) void k_conv1x1(
    const float* __restrict__ in, int bsi,
    const float* __restrict__ w, const float* __restrict__ bias,
    float* __restrict__ out, int bso,
    int Cin, int Cout, int M, int Bn, int nCoT,
    int sKin, int sMin, int sRow, int sCo) {
  const int lane = threadIdx.x & 31;
  const int wid = (blockIdx.x * blockDim.x + threadIdx.x) >> 5;
  const int posT = M >> 4;
  if (wid >= Bn * posT * nCoT) return;
  const int b  = wid / (posT * nCoT);
  const int rem = wid % (posT * nCoT);
  const int pos0 = (rem / nCoT) << 4;
  const int co0  = (rem % nCoT) << 4;
  const bool hi = lane >= 16;
  const int col = lane & 15;
  const int rowoff = hi ? 8 : 0;
  const int myco = co0 + col;
  const bool cov = myco < Cout;
  const int mycoC = cov ? myco : (Cout - 1);   // clamped; junk cols never stored
  const int m = pos0 + col;
  const float* inb = in + (size_t)b * bsi;
  const float* wrow = w + (size_t)mycoC * Cin;

  v8f acc = {};
  const int kMain = Cin & ~3;
  const int kofs = hi ? 2 : 0;
#pragma unroll 2
  for (int k0 = 0; k0 < kMain; k0 += 4) {
    const int ka = k0 + kofs;
    v2f af, bf;
    af.x = inb[(size_t)ka * sKin + (size_t)m * sMin];
    af.y = inb[(size_t)(ka + 1) * sKin + (size_t)m * sMin];
    bf.x = wrow[ka];
    bf.y = wrow[ka + 1];
    acc = __builtin_amdgcn_wmma_f32_16x16x4_f32(false, af, false, bf,
                                                (short)0, acc, false, false);
  }
  if (kMain < Cin) {                           // tail (Cin % 4 != 0): clamp + select
    const int ka = kMain + kofs;
    const int ka0 = (ka < Cin) ? ka : (Cin - 1);
    const int ka1 = (ka + 1 < Cin) ? (ka + 1) : (Cin - 1);
    v2f af, bf;
    float ax = inb[(size_t)ka0 * sKin + (size_t)m * sMin];
    float ay = inb[(size_t)ka1 * sKin + (size_t)m * sMin];
    af.x = (ka < Cin) ? ax : 0.f;
    af.y = (ka + 1 < Cin) ? ay : 0.f;
    float bx = wrow[ka0];
    float by = wrow[ka1];
    bf.x = (ka < Cin) ? bx : 0.f;
    bf.y = (ka + 1 < Cin) ? by : 0.f;
    acc = __builtin_amdgcn_wmma_f32_16x16x4_f32(false, af, false, bf,
                                                (short)0, acc, false, false);
  }
  const float bb = bias[mycoC];
  float* outb = out + (size_t)b * bso;
#pragma unroll
  for (int r = 0; r < 8; r++) {
    const int row = pos0 + rowoff + r;
    const float v = acc[r] + bb;
    if (cov) outb[(size_t)row * sRow + (size_t)myco * sCo] = v;
  }
}

// ------------------------------------------------------------- BatchNorm (train)
// mode 1: x is [B][C][M] with batch stride bstride; mode 0: row-major [M rows][C].
__global__ __launch_bounds__(256) void k_bn_reduce(
    const float* __restrict__ x, int bstride, int Bn, int C, int M, int mode,
    float* __restrict__ sums, float* __restrict__ sqs) {
  const int c = blockIdx.x;
  const int tid = threadIdx.x;
  float s = 0.f, qsum = 0.f;
  if (mode == 1) {
    long tot = (long)Bn * M;
    for (long idx = tid; idx < tot; idx += 256) {
      int b = (int)(idx / M);
      long mm = idx - (long)b * M;
      float v = x[(size_t)b * bstride + (size_t)c * M + mm];
      s += v; qsum += v * v;
    }
  } else {
    for (long r = tid; r < M; r += 256) {
      float v = x[(size_t)r * C + c];
      s += v; qsum += v * v;
    }
  }
  __shared__ float ss[256], sq[256];
  ss[tid] = s; sq[tid] = qsum;
  __syncthreads();
  for (int off = 128; off > 0; off >>= 1) {
    if (tid < off) { ss[tid] += ss[tid + off]; sq[tid] += sq[tid + off]; }
    __syncthreads();
  }
  if (tid == 0) { sums[c] = ss[0]; sqs[c] = sq[0]; }
}

__global__ __launch_bounds__(256) void k_bn_apply(
    float* __restrict__ x, int bstride, int Bn, int C, int M, int mode,
    const float* __restrict__ sums, const float* __restrict__ sqs,
    const float* __restrict__ g, const float* __restrict__ bt,
    float invcnt, int act) {
  long total = (mode == 1) ? (long)Bn * C * M : (long)C * M;
  for (long i = (long)blockIdx.x * blockDim.x + threadIdx.x; i < total;
       i += (long)gridDim.x * blockDim.x) {
    int c; size_t addr;
    if (mode == 1) {
      long cm = (long)C * M;
      int b = (int)(i / cm);
      long rem = i - (long)b * cm;
      c = (int)(rem / M);
      long mm = rem - (long)c * M;
      addr = (size_t)b * bstride + (size_t)c * M + mm;
    } else {
      c = (int)(i % C);
      addr = (size_t)i;
    }
    float mean = sums[c] * invcnt;
    float var  = sqs[c] * invcnt - mean * mean;
    float v = (x[addr] - mean) * rsqrtf(var + EPSV) * g[c] + bt[c];
    if (act == 1) v = fmaxf(v, 0.f);
    else if (act == 2) v = (v < 0.f) ? v * LEAK : v;
    x[addr] = v;
  }
}

// ------------------------------------------------------------- gathers / glue

__global__ void k_gather_nf(const float* __restrict__ f1, const int* __restrict__ neigh,
                            float* __restrict__ nfr) {
  long i = (long)blockIdx.x * blockDim.x + threadIdx.x;
  if (i >= (long)BB * 8 * KN) return;
  int b = (int)(i / ((long)8 * KN));
  long rem = i - (long)b * 8 * KN;
  long q = rem;                                 // cc*KN + kpos
  int j = (int)(q >> 3), ch = (int)(q & 7);
  int pt = neigh[(size_t)b * KN + j];
  nfr[i] = f1[(size_t)b * 8 * NP + (size_t)ch * NP + pt];
}

__global__ void k_feat_info(const float* __restrict__ f1, const float* __restrict__ nfr,
                            float* __restrict__ fi) {
  long i = (long)blockIdx.x * blockDim.x + threadIdx.x;
  if (i >= (long)BB * 8 * KN) return;
  int b = (int)(i / ((long)8 * KN));
  long rem = i - (long)b * 8 * KN;
  int cc = (int)(rem / KN);
  int kpos = (int)(rem % KN);
  int pos = kpos % NP;
  float nf = nfr[i];
  float tf = f1[(size_t)b * 8 * NP + (size_t)cc * NP + pos];
  fi[(size_t)b * 16 * KN + (size_t)cc * KN + kpos] = nf - tf;
  fi[(size_t)b * 16 * KN + (size_t)(8 + cc) * KN + kpos] = tf;
}

__global__ void k_gather_nx(const float* __restrict__ xyz, const int* __restrict__ neigh,
                            float* __restrict__ nxr) {
  long i = (long)blockIdx.x * blockDim.x + threadIdx.x;
  if (i >= (long)BB * 3 * KN) return;
  int b = (int)(i / ((long)3 * KN));
  long q = i - (long)b * 3 * KN;                // a*KN + kpos
  int j = (int)(q / 3), t = (int)(q % 3);
  int pt = neigh[(size_t)b * KN + j];
  nxr[i] = xyz[((size_t)b * NP + pt) * 3 + t];
}

__global__ void k_add(float* __restrict__ dst, const float* __restrict__ a, long n) {
  long i = (long)blockIdx.x * blockDim.x + threadIdx.x;
  if (i < n) dst[i] += a[i];
}

__global__ void k_xyz_info(const float* __restrict__ xyz, const float* __restrict__ nxr,
                           const float* __restrict__ shifted, float* __restrict__ xi) {
  long i = (long)blockIdx.x * blockDim.x + threadIdx.x;
  if (i >= (long)BB * 3 * KN) return;
  int b = (int)(i / ((long)3 * KN));
  long rem = i - (long)b * 3 * KN;
  int a = (int)(rem / KN);
  int kpos = (int)(rem % KN);
  int pos = kpos % NP;
  float tv = xyz[((size_t)b * NP + pos) * 3 + a];
  float nx = nxr[i];
  xi[(size_t)b * 9 * KN + (size_t)a * KN + kpos] = nx - tv;
  xi[(size_t)b * 9 * KN + (size_t)(3 + a) * KN + kpos] = shifted[i];
  xi[(size_t)b * 9 * KN + (size_t)(6 + a) * KN + kpos] = tv;
}

__global__ void k_fi2(const float* __restrict__ nfr, const float* __restrict__ nfo,
                      const float* __restrict__ fi, float* __restrict__ fi2) {
  long i = (long)blockIdx.x * blockDim.x + threadIdx.x;
  if (i >= (long)BB * 24 * KN) return;
  int b = (int)(i / ((long)24 * KN));
  long rem = i - (long)b * 24 * KN;
  int c = (int)(rem / KN);
  int kpos = (int)(rem % KN);
  if (c < 8) {
    size_t s = (size_t)b * 8 * KN + (size_t)c * KN + kpos;
    fi2[i] = nfr[s] + nfo[s];
  } else {
    fi2[i] = fi[(size_t)b * 16 * KN + (size_t)(c - 8) * KN + kpos];
  }
}

__global__ void k_softmax(float* __restrict__ kw) {
  long i = (long)blockIdx.x * blockDim.x + threadIdx.x;
  if (i >= (long)BB * 16 * NP) return;
  int b = (int)(i / ((long)16 * NP));
  long rem = i - (long)b * 16 * NP;
  int c = (int)(rem / NP);
  int pos = (int)(rem % NP);
  size_t base = ((size_t)b * 16 + c) * KN + pos;
  float m = -3.0e38f;
#pragma unroll
  for (int k = 0; k < KNB; k++) m = fmaxf(m, kw[base + (size_t)k * NP]);
  float ev[KNB]; float s = 0.f;
#pragma unroll
  for (int k = 0; k < KNB; k++) { ev[k] = expf(kw[base + (size_t)k * NP] - m); s += ev[k]; }
  float inv = 1.f / s;
#pragma unroll
  for (int k = 0; k < KNB; k++) kw[base + (size_t)k * NP] = ev[k] * inv;
}

__global__ void k_pool(const float* __restrict__ ov, const float* __restrict__ kw,
                       float* __restrict__ enc) {
  long i = (long)blockIdx.x * blockDim.x + threadIdx.x;
  if (i >= (long)BB * 16 * NP) return;
  int b = (int)(i / ((long)16 * NP));
  long rem = i - (long)b * 16 * NP;
  int c = (int)(rem / NP);
  int pos = (int)(rem % NP);
  size_t base = ((size_t)b * 16 + c) * KN + pos;
  float mx = -3.0e38f, ws = 0.f;
#pragma unroll
  for (int k = 0; k < KNB; k++) {
    float o = ov[base + (size_t)k * NP];
    mx = fmaxf(mx, o);
    ws += o * kw[base + (size_t)k * NP];
  }
  enc[(size_t)b * 32 * NP + (size_t)c * NP + pos] = mx;
  enc[(size_t)b * 32 * NP + (size_t)(16 + c) * NP + pos] = ws;
}

__global__ void k_fsamp(const float* __restrict__ fenc, const int* __restrict__ subi,
                        float* __restrict__ fs) {
  long i = (long)blockIdx.x * blockDim.x + threadIdx.x;
  if (i >= (long)BB * 32 * NSUB) return;
  int b = (int)(i / ((long)32 * NSUB));
  long rem = i - (long)b * 32 * NSUB;
  int c = (int)(rem / NSUB);
  int s = (int)(rem % NSUB);
  fs[i] = fenc[(size_t)b * 32 * NP + (size_t)c * NP + subi[(size_t)b * NSUB + s]];
}

__global__ void k_ip(const float* __restrict__ fenc, const float* __restrict__ mfeat,
                     const int* __restrict__ itp, float* __restrict__ ip) {
  long i = (long)blockIdx.x * blockDim.x + threadIdx.x;
  if (i >= (long)BB * 64 * NP) return;
  int b = (int)(i / ((long)64 * NP));
  long rem = i - (long)b * 64 * NP;
  int c = (int)(rem / NP);
  int pos = (int)(rem % NP);
  if (c < 32) {
    ip[i] = fenc[(size_t)b * 32 * NP + (size_t)c * NP + pos];
  } else {
    int pt = itp[(size_t)b * NP + pos];
    ip[i] = mfeat[(size_t)b * 32 * NSUB + (size_t)(c - 32) * NSUB + pt];
  }
}

// ------------------------------------------------------------- host helpers

inline int gblk(long total) { return (int)((total + 255) / 256); }

void run_conv(hipStream_t st, const float* in, int bsi, const float* w, const float* bias,
              float* out, int bso, int Cin, int Cout, int M, int Bn, int inRM, int outRM) {
  int nCoT = (Cout + 15) / 16;
  long waves = (long)Bn * (M / 16) * nCoT;
  int blocks = (int)((waves * 32 + 255) / 256);
  int sKin = inRM ? 1 : M;
  int sMin = inRM ? Cin : 1;
  int sRow = outRM ? Cout : 1;
  int sCo  = outRM ? 1 : M;
  k_conv1x1<<<blocks, 256, 0, st>>>(in, bsi, w, bias, out, bso, Cin, Cout, M, Bn, nCoT,
                                    sKin, sMin, sRow, sCo);
}

void run_bn(hipStream_t st, float* xb, int bstride, int Bn, int C, int M, int mode,
            const float* g, const float* bt, int act, float* sums, float* sqs, long count) {
  k_bn_reduce<<<C, 256, 0, st>>>(xb, bstride, Bn, C, M, mode, sums, sqs);
  long total = (mode == 1) ? (long)Bn * C * M : (long)C * M;
  int blocks = (int)((total + 255) / 256);
  if (blocks > 4096) blocks = 4096;
  k_bn_apply<<<blocks, 256, 0, st>>>(xb, bstride, Bn, C, M, mode, sums, sqs, g, bt,
                                     1.0f / (float)count, act);
}

}  // namespace

extern "C" void kernel_launch(void* const* d_in, const int* in_sizes, int n_in,
                              void* d_out, int out_size, void* d_ws, size_t ws_size,
                              hipStream_t stream) {
  (void)in_sizes; (void)n_in; (void)out_size; (void)ws_size;
  const float* x = (const float*)d_in[0];
  const float* fc0_w = (const float*)d_in[1];  const float* fc0_b = (const float*)d_in[2];
  const float* bn0_g = (const float*)d_in[3];  const float* bn0_b = (const float*)d_in[4];
  const float* c1_w  = (const float*)d_in[5];  const float* c1_b  = (const float*)d_in[6];
  const float* bn1_g = (const float*)d_in[7];  const float* bn1_b = (const float*)d_in[8];
  const float* c2_w  = (const float*)d_in[9];  const float* c2_b  = (const float*)d_in[10];
  const float* bn2_g = (const float*)d_in[11]; const float* bn2_b = (const float*)d_in[12];
  const float* c3_w  = (const float*)d_in[13]; const float* c3_b  = (const float*)d_in[14];
  const float* bn3_g = (const float*)d_in[15]; const float* bn3_b = (const float*)d_in[16];
  const float* c4_w  = (const float*)d_in[17]; const float* c4_b  = (const float*)d_in[18];
  const float* bn4_g = (const float*)d_in[19]; const float* bn4_b = (const float*)d_in[20];
  const float* c5_w  = (const float*)d_in[21]; const float* c5_b  = (const float*)d_in[22];
  const float* bn5_g = (const float*)d_in[23]; const float* bn5_b = (const float*)d_in[24];
  const float* c6_w  = (const float*)d_in[25]; const float* c6_b  = (const float*)d_in[26];
  const float* c7_w  = (const float*)d_in[27]; const float* c7_b  = (const float*)d_in[28];
  const float* bn7_g = (const float*)d_in[29]; const float* bn7_b = (const float*)d_in[30];
  const float* c8_w  = (const float*)d_in[31]; const float* c8_b  = (const float*)d_in[32];
  const float* bn8_g = (const float*)d_in[33]; const float* bn8_b = (const float*)d_in[34];
  const float* m1_w  = (const float*)d_in[35]; const float* m1_b  = (const float*)d_in[36];
  const float* mbn1_g= (const float*)d_in[37]; const float* mbn1_b= (const float*)d_in[38];
  const float* up1_w = (const float*)d_in[39]; const float* up1_b = (const float*)d_in[40];
  const float* ubn1_g= (const float*)d_in[41]; const float* ubn1_b= (const float*)d_in[42];
  const float* m3_w  = (const float*)d_in[43]; const float* m3_b  = (const float*)d_in[44];
  const float* mbn3_g= (const float*)d_in[45]; const float* mbn3_b= (const float*)d_in[46];
  const float* m4_w  = (const float*)d_in[47]; const float* m4_b  = (const float*)d_in[48];
  const float* mbn4_g= (const float*)d_in[49]; const float* mbn4_b= (const float*)d_in[50];
  const float* m5_w  = (const float*)d_in[51]; const float* m5_b  = (const float*)d_in[52];
  float* outp = (float*)d_out;

  // ---- workspace carve (bump allocator, 256B aligned)
  char* wsb = (char*)d_ws;
  size_t off = 0;
  auto alf = [&](size_t elems) -> float* {
    float* p = (float*)(wsb + off);
    off += ((elems * sizeof(float) + 255) / 256) * 256;
    return p;
  };
  auto ali = [&](size_t elems) -> int* {
    int* p = (int*)(wsb + off);
    off += ((elems * sizeof(int) + 255) / 256) * 256;
    return p;
  };
  float* xyz   = alf((size_t)BB * NP * 3);
  float* qn    = alf((size_t)BB * NP);
  float* sxyz  = alf((size_t)BB * NSUB * 3);
  float* f0    = alf((size_t)BB * NP * 32);
  float* f1    = alf((size_t)BB * 8 * NP);
  float* nfr   = alf((size_t)BB * 8 * KN);
  float* fi    = alf((size_t)BB * 16 * KN);
  float* nxr   = alf((size_t)BB * 3 * KN);
  float* nxo   = alf((size_t)BB * 3 * KN);
  float* xi    = alf((size_t)BB * 9 * KN);
  float* nfo   = alf((size_t)BB * 8 * KN);
  float* fi2   = alf((size_t)BB * 24 * KN);
  float* ov    = alf((size_t)BB * 16 * KN);
  float* kwb   = alf((size_t)BB * 16 * KN);
  float* enc   = alf((size_t)BB * 32 * NP);
  float* enc2  = alf((size_t)BB * 16 * NP);
  float* fenc  = alf((size_t)BB * 32 * NP);
  float* fsb   = alf((size_t)BB * 32 * NSUB);
  float* mfeat = alf((size_t)BB * 32 * NSUB);
  float* ipb   = alf((size_t)BB * 64 * NP);
  float* dec   = alf((size_t)BB * 32 * NP);
  float* h1    = alf((size_t)BB * 64 * NP);
  float* h2    = alf((size_t)BB * 32 * NP);
  float* sums  = alf(64);
  float* sqs   = alf(64);
  int* neigh   = ali((size_t)BB * NP * KNB);
  int* subi    = ali((size_t)BB * NSUB);
  int* itp     = ali((size_t)BB * NP);

  hipStream_t st = stream;

  // ---- geometry / indices
  k_build_xyz<<<gblk((long)BB * NP * 3), 256, 0, st>>>(x, xyz);
  k_norms<<<gblk((long)BB * NP), 256, 0, st>>>(xyz, qn);
  k_knn<<<(BB * (NP / 16) + 3) / 4, 128, 0, st>>>(xyz, qn, neigh);
  k_fps<<<BB, 1024, 0, st>>>(xyz, subi);
  k_subxyz<<<gblk((long)BB * NSUB), 256, 0, st>>>(xyz, subi, sxyz);
  k_interp<<<BB * NP / 256, 256, 0, st>>>(xyz, sxyz, itp);

  // ---- fc0 + bn1d (leaky)
  run_conv(st, x, 0, fc0_w, fc0_b, f0, 0, 24, 32, BB * NP, 1, 1, 1);
  run_bn(st, f0, 0, 1, 32, BB * NP, 0, bn0_g, bn0_b, 2, sums, sqs, (long)BB * NP);

  // ---- bilateral context block
  run_conv(st, f0, 32 * NP, c1_w, c1_b, f1, 8 * NP, 32, 8, NP, BB, 0, 0);
  run_bn(st, f1, 8 * NP, BB, 8, NP, 1, bn1_g, bn1_b, 1, sums, sqs, (long)BB * NP);

  k_gather_nf<<<gblk((long)BB * 8 * KN), 256, 0, st>>>(f1, neigh, nfr);
  k_feat_info<<<gblk((long)BB * 8 * KN), 256, 0, st>>>(f1, nfr, fi);
  k_gather_nx<<<gblk((long)BB * 3 * KN), 256, 0, st>>>(xyz, neigh, nxr);

  run_conv(st, fi, 16 * KN, c2_w, c2_b, nxo, 3 * KN, 16, 3, KN, BB, 0, 0);
  run_bn(st, nxo, 3 * KN, BB, 3, KN, 1, bn2_g, bn2_b, 1, sums, sqs, (long)BB * KN);
  k_add<<<gblk((long)BB * 3 * KN), 256, 0, st>>>(nxo, nxr, (long)BB * 3 * KN);  // shifted_xyz
  k_xyz_info<<<gblk((long)BB * 3 * KN), 256, 0, st>>>(xyz, nxr, nxo, xi);

  run_conv(st, xi, 9 * KN, c3_w, c3_b, nfo, 8 * KN, 9, 8, KN, BB, 0, 0);
  run_bn(st, nfo, 8 * KN, BB, 8, KN, 1, bn3_g, bn3_b, 1, sums, sqs, (long)BB * KN);

  run_conv(st, xi, 9 * KN, c4_w, c4_b, ov, 16 * KN, 9, 8, KN, BB, 0, 0);  // xyz_enc -> ov[0:8]
  run_bn(st, ov, 16 * KN, BB, 8, KN, 1, bn4_g, bn4_b, 1, sums, sqs, (long)BB * KN);

  k_fi2<<<gblk((long)BB * 24 * KN), 256, 0, st>>>(nfr, nfo, fi, fi2);
  run_conv(st, fi2, 24 * KN, c5_w, c5_b, ov + (size_t)8 * KN, 16 * KN, 24, 8, KN, BB, 0, 0);
  run_bn(st, ov + (size_t)8 * KN, 16 * KN, BB, 8, KN, 1, bn5_g, bn5_b, 1, sums, sqs,
         (long)BB * KN);

  run_conv(st, ov, 16 * KN, c6_w, c6_b, kwb, 16 * KN, 16, 16, KN, BB, 0, 0);
  k_softmax<<<gblk((long)BB * 16 * NP), 256, 0, st>>>(kwb);
  k_pool<<<gblk((long)BB * 16 * NP), 256, 0, st>>>(ov, kwb, enc);

  run_conv(st, enc, 32 * NP, c7_w, c7_b, enc2, 16 * NP, 32, 16, NP, BB, 0, 0);
  run_bn(st, enc2, 16 * NP, BB, 16, NP, 1, bn7_g, bn7_b, 1, sums, sqs, (long)BB * NP);
  run_conv(st, enc2, 16 * NP, c8_w, c8_b, fenc, 32 * NP, 16, 32, NP, BB, 0, 0);
  run_bn(st, fenc, 32 * NP, BB, 32, NP, 1, bn8_g, bn8_b, 2, sums, sqs, (long)BB * NP);

  // ---- downsample / upsample path
  k_fsamp<<<gblk((long)BB * 32 * NSUB), 256, 0, st>>>(fenc, subi, fsb);
  run_conv(st, fsb, 32 * NSUB, m1_w, m1_b, mfeat, 32 * NSUB, 32, 32, NSUB, BB, 0, 0);
  run_bn(st, mfeat, 32 * NSUB, BB, 32, NSUB, 1, mbn1_g, mbn1_b, 2, sums, sqs,
         (long)BB * NSUB);

  k_ip<<<gblk((long)BB * 64 * NP), 256, 0, st>>>(fenc, mfeat, itp, ipb);
  run_conv(st, ipb, 64 * NP, up1_w, up1_b, dec, 32 * NP, 64, 32, NP, BB, 0, 0);
  run_bn(st, dec, 32 * NP, BB, 32, NP, 1, ubn1_g, ubn1_b, 2, sums, sqs, (long)BB * NP);

  // ---- head
  run_conv(st, dec, 32 * NP, m3_w, m3_b, h1, 64 * NP, 32, 64, NP, BB, 0, 0);
  run_bn(st, h1, 64 * NP, BB, 64, NP, 1, mbn3_g, mbn3_b, 1, sums, sqs, (long)BB * NP);
  run_conv(st, h1, 64 * NP, m4_w, m4_b, h2, 32 * NP, 64, 32, NP, BB, 0, 0);
  run_bn(st, h2, 32 * NP, BB, 32, NP, 1, mbn4_g, mbn4_b, 1, sums, sqs, (long)BB * NP);
  run_conv(st, h2, 32 * NP, m5_w, m5_b, outp, NCLS * NP, 32, NCLS, NP, BB, 0, 0);
}